// HeteroGATEncoder_6760278524232
// MI455X (gfx1250) — compile-verified
//
#include <hip/hip_runtime.h>
#include <hip/hip_bf16.h>
#include <math.h>

#define N_USERS 50000
#define N_ITEMS 100000
#define E_EDGES 400000
#define HIDD    128
#define OUTD    64
#define HEADS   4
#define NEG_SLOPE 0.2f
#define LN_EPS  1e-5f

typedef __attribute__((ext_vector_type(2))) float v2f;
typedef __attribute__((ext_vector_type(8))) float v8f;

// ---------- float <-> order-preserving int (for atomic max on floats) ----------
__device__ __forceinline__ int f2o(float f) {
    int i = __float_as_int(f);
    return (i >= 0) ? i : (i ^ 0x7fffffff);
}
__device__ __forceinline__ float o2f(int i) {
    return __int_as_float((i >= 0) ? i : (i ^ 0x7fffffff));
}

// =====================================================================
// GEMM: C[N,M] = A[N,K] @ W[K,M] + bias[M]   (fp32 WMMA 16x16x4)
//
// Block = 128 threads (4 waves). Block owns a K x 64 strip of W, staged
// once into LDS *transposed* ([64][K+2], +2 pad keeps ds_load_b64 8B-aligned
// and gives bank stride 2 -> conflict-free wave32 fill). Each wave computes
// a 16x64 tile: per K-step = 1 global b64 (A frag) + 4 ds_load_b64 (B frags)
// + 4 WMMAs on 4 independent accumulator chains.
//
// Requires N%16==0, M%64==0, K%4==0 (true: N in {50000,100000},
// M in {128,64}, K in {128,256}).
//
// A frag: lane L holds A[row0 + (L&15), k + 2*(L>>4) + {0,1}]
// B frag: lane L holds W[k + 2*(L>>4) + {0,1}, col + (L&15)]
// D: vgpr r, lane L -> C[row0 + r + 8*(L>>4), col + (L&15)]
// =====================================================================
__global__ void gemm_bias_wmma(const float* __restrict__ A,
                               const float* __restrict__ W,
                               const float* __restrict__ bias,
                               float* __restrict__ C,
                               int N, int K, int M) {
    extern __shared__ float wlds[];          // [64][K+2] transposed W strip
    const int KP   = K + 2;
    const int wave = threadIdx.x >> 5;
    const int lane = threadIdx.x & 31;
    const int row0 = (blockIdx.x * 4 + wave) * 16;
    const int col0 = blockIdx.y * 64;

    // ---- cooperative fill: wlds[c][k] = W[k][col0+c] (coalesced global) ----
    {
        const float* wp = W + col0;
        for (int i = threadIdx.x; i < K * 64; i += 128) {
            int k = i >> 6;                  // 0..K-1
            int c = i & 63;                  // 0..63
            wlds[c * KP + k] = wp[(unsigned)k * (unsigned)M + c];
        }
    }
    __syncthreads();

    if (row0 < N) {                          // wave-uniform: EXEC all-1s inside
        const int m    = lane & 15;
        const int koff = (lane >> 4) << 1;

        v8f acc[4];
#pragma unroll
        for (int t = 0; t < 4; ++t)
            acc[t] = (v8f){0.f, 0.f, 0.f, 0.f, 0.f, 0.f, 0.f, 0.f};

        // 32-bit offsets: worst case (row0+m)*K < 100000*256 = 25.6M < 2^31
        const float* ap = A + (unsigned)(row0 + m) * (unsigned)K + koff;
        const float* lb0 = wlds + (0 * 16 + m) * KP + koff;
        const float* lb1 = wlds + (1 * 16 + m) * KP + koff;
        const float* lb2 = wlds + (2 * 16 + m) * KP + koff;
        const float* lb3 = wlds + (3 * 16 + m) * KP + koff;

#pragma unroll 4
        for (int k = 0; k < K; k += 4) {
            v2f a;
            a.x = ap[k];
            a.y = ap[k + 1];
            v2f b0 = *(const v2f*)(lb0 + k);   // ds_load_b64 each
            v2f b1 = *(const v2f*)(lb1 + k);
            v2f b2 = *(const v2f*)(lb2 + k);
            v2f b3 = *(const v2f*)(lb3 + k);
            acc[0] = __builtin_amdgcn_wmma_f32_16x16x4_f32(
                false, a, false, b0, (short)0, acc[0], false, false);
            acc[1] = __builtin_amdgcn_wmma_f32_16x16x4_f32(
                false, a, false, b1, (short)0, acc[1], false, false);
            acc[2] = __builtin_amdgcn_wmma_f32_16x16x4_f32(
                false, a, false, b2, (short)0, acc[2], false, false);
            acc[3] = __builtin_amdgcn_wmma_f32_16x16x4_f32(
                false, a, false, b3, (short)0, acc[3], false, false);
        }

        const int n     = lane & 15;
        const int mbase = (lane >> 4) * 8;
#pragma unroll
        for (int t = 0; t < 4; ++t) {
            const int col = col0 + t * 16 + n;
            const float bv = bias[col];
            float* cp = C + (unsigned)(row0 + mbase) * (unsigned)M + col;
#pragma unroll
            for (int r = 0; r < 8; ++r) {
                cp[(unsigned)r * (unsigned)M] = acc[t][r] + bv;
            }
        }
    }
}

// =====================================================================
// Init destination-side buffers: out[n,:] = gat_bias, denom = 0, smax = f2o(-inf)
// =====================================================================
__global__ void init_dst_kernel(float* __restrict__ out, float* __restrict__ denom,
                                int* __restrict__ smax, const float* __restrict__ bias,
                                int n_dst, int dim) {
    int i = blockIdx.x * blockDim.x + threadIdx.x;
    int total = n_dst * dim;
    if (i < total) out[i] = bias[i & (dim - 1)];
    if (i < n_dst * HEADS) {
        denom[i] = 0.f;
        smax[i]  = f2o(-INFINITY);
    }
}

// =====================================================================
// Pass 1: per-edge, per-head score = att . leaky_relu(xl[src]+xr[dst]),
// store score, atomicMax into per-(dst,head) running max.
// One wave per edge; 8 lanes per head; lane covers DIM/32 contiguous elems.
// =====================================================================
template <int DIM>
__global__ void edge_score_kernel(const float* __restrict__ xl, const float* __restrict__ xr,
                                  const float* __restrict__ att,
                                  const int* __restrict__ src, const int* __restrict__ dst,
                                  float* __restrict__ score, int* __restrict__ smax, int E) {
    constexpr int VEC = DIM / 32;
    int e = blockIdx.x * (blockDim.x >> 5) + (threadIdx.x >> 5);
    if (e >= E) return;
    int lane = threadIdx.x & 31;
    int s = src[e], d = dst[e];
    const float* pl = xl + (unsigned)s * DIM + lane * VEC;   // < 12.8M, fits 32-bit
    const float* pr = xr + (unsigned)d * DIM + lane * VEC;
    const float* pa = att + lane * VEC;
    float partial = 0.f;
#pragma unroll
    for (int j = 0; j < VEC; ++j) {
        float v = pl[j] + pr[j];
        v = (v > 0.f) ? v : v * NEG_SLOPE;
        partial += v * pa[j];
    }
    // reduce over the 8 lanes of this head
    partial += __shfl_xor(partial, 1, 32);
    partial += __shfl_xor(partial, 2, 32);
    partial += __shfl_xor(partial, 4, 32);
    if ((lane & 7) == 0) {
        int h = lane >> 3;
        score[(unsigned)e * HEADS + h] = partial;
        atomicMax(&smax[d * HEADS + h], f2o(partial));
    }
}

// =====================================================================
// Pass 2: a = exp(score - max[dst]); score <- a; denom[dst,h] += a
// One thread per (edge, head).
// =====================================================================
__global__ void edge_exp_kernel(const int* __restrict__ dst, const int* __restrict__ smax,
                                float* __restrict__ score, float* __restrict__ denom, int E4) {
    int i = blockIdx.x * blockDim.x + threadIdx.x;
    if (i >= E4) return;
    int e = i >> 2;
    int h = i & 3;
    int d = dst[e];
    float m = o2f(smax[d * HEADS + h]);
    float a = __expf(score[i] - m);
    score[i] = a;
    atomicAdd(&denom[d * HEADS + h], a);
}

// =====================================================================
// Pass 3: out[dst] += (a / denom[dst]) * xl[src]   (per-head weight)
// One wave per edge.
// =====================================================================
template <int DIM>
__global__ void edge_aggregate_kernel(const float* __restrict__ xl, const float* __restrict__ a,
                                      const float* __restrict__ denom,
                                      const int* __restrict__ src, const int* __restrict__ dst,
                                      float* __restrict__ out, int E) {
    constexpr int VEC = DIM / 32;
    int e = blockIdx.x * (blockDim.x >> 5) + (threadIdx.x >> 5);
    if (e >= E) return;
    int lane = threadIdx.x & 31;
    int s = src[e], d = dst[e];
    int h = lane >> 3;
    float w = a[(unsigned)e * HEADS + h] / (denom[d * HEADS + h] + 1e-16f);
    const float* pl = xl + (unsigned)s * DIM + lane * VEC;
    float* po = out + (unsigned)d * DIM + lane * VEC;
#pragma unroll
    for (int j = 0; j < VEC; ++j) atomicAdd(&po[j], w * pl[j]);
}

// =====================================================================
// LayerNorm (+ optional residual, optional ELU). One wave per row.
// =====================================================================
template <int DIM, bool RESID, bool ELU>
__global__ void ln_kernel(const float* __restrict__ x, const float* __restrict__ resid,
                          const float* __restrict__ g, const float* __restrict__ b,
                          float* __restrict__ out, int N) {
    constexpr int VEC = DIM / 32;
    int row = blockIdx.x * (blockDim.x >> 5) + (threadIdx.x >> 5);
    if (row >= N) return;
    int lane = threadIdx.x & 31;
    const float* px = x + (unsigned)row * DIM + lane * VEC;
    float v[VEC];
    float s = 0.f;
#pragma unroll
    for (int j = 0; j < VEC; ++j) { v[j] = px[j]; s += v[j]; }
#pragma unroll
    for (int m = 1; m < 32; m <<= 1) s += __shfl_xor(s, m, 32);
    float mu = s * (1.0f / DIM);
    float q = 0.f;
#pragma unroll
    for (int j = 0; j < VEC; ++j) { float d0 = v[j] - mu; q += d0 * d0; }
#pragma unroll
    for (int m = 1; m < 32; m <<= 1) q += __shfl_xor(q, m, 32);
    float rstd = rsqrtf(q * (1.0f / DIM) + LN_EPS);

    float* po = out + (unsigned)row * DIM + lane * VEC;
    const float* pr = resid + (unsigned)row * DIM + lane * VEC;
#pragma unroll
    for (int j = 0; j < VEC; ++j) {
        int c = lane * VEC + j;
        float y = (v[j] - mu) * rstd * g[c] + b[c];
        if (RESID) y += pr[j];
        if (ELU) y = (y > 0.f) ? y : expm1f(y);
        po[j] = y;
    }
}

// =====================================================================
// Host orchestration
// =====================================================================
extern "C" void kernel_launch(void* const* d_in, const int* in_sizes, int n_in,
                              void* d_out, int out_size, void* d_ws, size_t ws_size,
                              hipStream_t stream) {
    // --- input index mapping ---------------------------------------------
    // Default: top-level dict insertion order, params flattened with jax
    // pytree (sorted dict keys):
    //   per layer (base = li*16):
    //     +0 iu.Wl  +1 iu.Wr  +2 iu.att  +3 iu.bias  +4 iu.bl  +5 iu.br
    //     +6 ln.item.b +7 ln.item.g +8 ln.user.b +9 ln.user.g
    //     +10 ui.Wl +11 ui.Wr +12 ui.att +13 ui.bias +14 ui.bl +15 ui.br
    //   48 proj.item.W  49 proj.item.b  50 proj.user.W  51 proj.user.b
    //   52 x_user  53 x_item  54 ei_ui  55 ei_iu
    // Fallback (whole dict sorted): ei_iu(0), ei_ui(1), layer leaves (2..49),
    //   proj (50..53), x_item(54), x_user(55).
    int pbase, iproj, ix_user, ix_item, iei_ui, iei_iu;
    if (n_in >= 1 && in_sizes[0] == 2 * E_EDGES) {
        iei_iu = 0; iei_ui = 1; pbase = 2; iproj = 50; ix_item = 54; ix_user = 55;
    } else {
        pbase = 0; iproj = 48; ix_user = 52; ix_item = 53; iei_ui = 54; iei_iu = 55;
    }
    auto F = [&](int i) { return (const float*)d_in[i]; };
    const float* x_user = F(ix_user);
    const float* x_item = F(ix_item);
    const int* ei_ui = (const int*)d_in[iei_ui];   // [2,E]: row0 = user(src), row1 = item(dst)
    const int* ei_iu = (const int*)d_in[iei_iu];   // [2,E]: row0 = item(src), row1 = user(dst)

    // --- workspace carve-out ----------------------------------------------
    float* ws = (float*)d_ws;
    size_t off = 0;
    auto alloc = [&](size_t n) { float* p = ws + off; off += n; return p; };
    float* hu       = alloc((size_t)N_USERS * HIDD);
    float* hi       = alloc((size_t)N_ITEMS * HIDD);
    float* xl_ui    = alloc((size_t)N_USERS * HIDD);
    float* xr_ui    = alloc((size_t)N_ITEMS * HIDD);
    float* xl_iu    = alloc((size_t)N_ITEMS * HIDD);
    float* xr_iu    = alloc((size_t)N_USERS * HIDD);
    float* new_u    = alloc((size_t)N_USERS * HIDD);
    float* new_i    = alloc((size_t)N_ITEMS * HIDD);
    float* score_ui = alloc((size_t)E_EDGES * HEADS);
    float* score_iu = alloc((size_t)E_EDGES * HEADS);
    float* den_u    = alloc((size_t)N_USERS * HEADS);
    float* den_i    = alloc((size_t)N_ITEMS * HEADS);
    int*   smax_u   = (int*)alloc((size_t)N_USERS * HEADS);
    int*   smax_i   = (int*)alloc((size_t)N_ITEMS * HEADS);
    (void)ws_size;

    auto gemm = [&](const float* A, const float* W, const float* bias, float* C,
                    int N, int K, int M) {
        dim3 grid((N / 16 + 3) / 4, M / 64);
        size_t ldsBytes = (size_t)(K + 2) * 64 * sizeof(float);
        gemm_bias_wmma<<<grid, 128, ldsBytes, stream>>>(A, W, bias, C, N, K, M);
    };

    const int EW = (E_EDGES + 7) / 8;   // wave-per-edge grids (8 waves / block)

    // --- initial projections ----------------------------------------------
    gemm(x_user, F(iproj + 2), F(iproj + 3), hu, N_USERS, 128, HIDD);   // user.W/b
    gemm(x_item, F(iproj + 0), F(iproj + 1), hi, N_ITEMS, 256, HIDD);   // item.W/b

    for (int li = 0; li < 3; ++li) {
        const int base = pbase + li * 16;
        const int dim  = (li < 2) ? HIDD : OUTD;
        const float *iuWl = F(base + 0),  *iuWr = F(base + 1),  *iuAtt = F(base + 2);
        const float *iuBias = F(base + 3), *iuBl = F(base + 4), *iuBr = F(base + 5);
        const float *lnIb = F(base + 6), *lnIg = F(base + 7);
        const float *lnUb = F(base + 8), *lnUg = F(base + 9);
        const float *uiWl = F(base + 10), *uiWr = F(base + 11), *uiAtt = F(base + 12);
        const float *uiBias = F(base + 13), *uiBl = F(base + 14), *uiBr = F(base + 15);

        // projections for both directions (all read pre-update hu/hi)
        gemm(hu, uiWl, uiBl, xl_ui, N_USERS, HIDD, dim);
        gemm(hi, uiWr, uiBr, xr_ui, N_ITEMS, HIDD, dim);
        gemm(hi, iuWl, iuBl, xl_iu, N_ITEMS, HIDD, dim);
        gemm(hu, iuWr, iuBr, xr_iu, N_USERS, HIDD, dim);

        const int* src_ui = ei_ui;              const int* dst_ui = ei_ui + E_EDGES;
        const int* src_iu = ei_iu;              const int* dst_iu = ei_iu + E_EDGES;

        init_dst_kernel<<<(N_ITEMS * dim + 255) / 256, 256, 0, stream>>>(
            new_i, den_i, smax_i, uiBias, N_ITEMS, dim);
        init_dst_kernel<<<(N_USERS * dim + 255) / 256, 256, 0, stream>>>(
            new_u, den_u, smax_u, iuBias, N_USERS, dim);

        if (dim == HIDD) {
            edge_score_kernel<HIDD><<<EW, 256, 0, stream>>>(
                xl_ui, xr_ui, uiAtt, src_ui, dst_ui, score_ui, smax_i, E_EDGES);
            edge_score_kernel<HIDD><<<EW, 256, 0, stream>>>(
                xl_iu, xr_iu, iuAtt, src_iu, dst_iu, score_iu, smax_u, E_EDGES);
        } else {
            edge_score_kernel<OUTD><<<EW, 256, 0, stream>>>(
                xl_ui, xr_ui, uiAtt, src_ui, dst_ui, score_ui, smax_i, E_EDGES);
            edge_score_kernel<OUTD><<<EW, 256, 0, stream>>>(
                xl_iu, xr_iu, iuAtt, src_iu, dst_iu, score_iu, smax_u, E_EDGES);
        }

        edge_exp_kernel<<<(E_EDGES * HEADS + 255) / 256, 256, 0, stream>>>(
            dst_ui, smax_i, score_ui, den_i, E_EDGES * HEADS);
        edge_exp_kernel<<<(E_EDGES * HEADS + 255) / 256, 256, 0, stream>>>(
            dst_iu, smax_u, score_iu, den_u, E_EDGES * HEADS);

        if (dim == HIDD) {
            edge_aggregate_kernel<HIDD><<<EW, 256, 0, stream>>>(
                xl_ui, score_ui, den_i, src_ui, dst_ui, new_i, E_EDGES);
            edge_aggregate_kernel<HIDD><<<EW, 256, 0, stream>>>(
                xl_iu, score_iu, den_u, src_iu, dst_iu, new_u, E_EDGES);
        } else {
            edge_aggregate_kernel<OUTD><<<EW, 256, 0, stream>>>(
                xl_ui, score_ui, den_i, src_ui, dst_ui, new_i, E_EDGES);
            edge_aggregate_kernel<OUTD><<<EW, 256, 0, stream>>>(
                xl_iu, score_iu, den_u, src_iu, dst_iu, new_u, E_EDGES);
        }

        // LayerNorm (+residual +ELU for layers 0,1); final layer -> d_out
        if (li < 2) {
            ln_kernel<HIDD, true, true><<<(N_USERS + 7) / 8, 256, 0, stream>>>(
                new_u, hu, lnUg, lnUb, hu, N_USERS);
            ln_kernel<HIDD, true, true><<<(N_ITEMS + 7) / 8, 256, 0, stream>>>(
                new_i, hi, lnIg, lnIb, hi, N_ITEMS);
        } else {
            float* out_u = (float*)d_out;
            float* out_i = (float*)d_out + (size_t)N_USERS * OUTD;
            ln_kernel<OUTD, false, false><<<(N_USERS + 7) / 8, 256, 0, stream>>>(
                new_u, new_u, lnUg, lnUb, out_u, N_USERS);
            ln_kernel<OUTD, false, false><<<(N_ITEMS + 7) / 8, 256, 0, stream>>>(
                new_i, new_i, lnIg, lnIb, out_i, N_ITEMS);
        }
    }
    (void)in_sizes; (void)n_in; (void)out_size;
}